// ConstantVelocityModel_16277926052562
// MI455X (gfx1250) — compile-verified
//
#include <hip/hip_runtime.h>
#include <math.h>

typedef float v2f __attribute__((ext_vector_type(2)));
typedef float v8f __attribute__((ext_vector_type(8)));

// ---------------------------------------------------------------------------
// Wave-level FP32 sum via V_WMMA_F32_16X16X4_F32 (wave32).
// A-matrix 16x4 f32 layout: VGPR0 lanes0-15 = A[m][0], lanes16-31 = A[m][2];
// VGPR1 = K=1/K=3 (zeroed).  B = ones(4x16)  =>  D[m][n] = s_m + s_{m+16}.
// Second stage: A = ones(16x4), B built from D VGPR pairs, C-accumulated
// over 4 WMMAs => every element of C holds sum over all 32 lanes (pure FP32).
// ---------------------------------------------------------------------------
__device__ __forceinline__ float wave_reduce_wmma(float s) {
  v2f a;   a[0] = s;     a[1] = 0.0f;
  v2f one; one[0] = 1.0f; one[1] = 1.0f;
  v8f z = {};
  v8f d = __builtin_amdgcn_wmma_f32_16x16x4_f32(false, a, false, one,
                                                (short)0, z, false, false);
  v8f c = {};
  v2f b;
  b[0] = d[0]; b[1] = d[1];
  c = __builtin_amdgcn_wmma_f32_16x16x4_f32(false, one, false, b,
                                            (short)0, c, false, false);
  b[0] = d[2]; b[1] = d[3];
  c = __builtin_amdgcn_wmma_f32_16x16x4_f32(false, one, false, b,
                                            (short)0, c, false, false);
  b[0] = d[4]; b[1] = d[5];
  c = __builtin_amdgcn_wmma_f32_16x16x4_f32(false, one, false, b,
                                            (short)0, c, false, false);
  b[0] = d[6]; b[1] = d[7];
  c = __builtin_amdgcn_wmma_f32_16x16x4_f32(false, one, false, b,
                                            (short)0, c, false, false);
  return c[0];   // total replicated across all lanes / elements
}

// Block reduction (blockDim.x multiple of 32, <= 1024). Result valid in wave 0.
__device__ __forceinline__ float block_reduce(float s) {
  __shared__ float wpart[32];
  const int lane = threadIdx.x & 31;
  const int wid  = threadIdx.x >> 5;
  const int nw   = blockDim.x >> 5;

  float ws = wave_reduce_wmma(s);          // full EXEC here (uniform path)
  if (lane == 0) wpart[wid] = ws;
  __syncthreads();

  float r = 0.0f;
  if (wid == 0) {                          // whole wave 0 enters: EXEC all-1s
    float v = (lane < nw) ? wpart[lane] : 0.0f;
    r = wave_reduce_wmma(v);
  }
  __syncthreads();                         // protect wpart for back-to-back use
  return r;
}

// ---------------------------------------------------------------------------
// Kernel 1: event intensity partials.  sum_e || (z_i - z_j) + (v_i - v_j) t ||^2
// 12 MB coalesced stream + cached 8B gathers from 40KB tables.
// ---------------------------------------------------------------------------
__global__ void cvm_event_kernel(const int2* __restrict__ idx,
                                 const float* __restrict__ t,
                                 const float2* __restrict__ z0,
                                 const float2* __restrict__ v0,
                                 int n_events,
                                 float* __restrict__ partials) {
  float s = 0.0f;
  const int stride = gridDim.x * blockDim.x;
  for (int e = blockIdx.x * blockDim.x + threadIdx.x; e < n_events; e += stride) {
    const int2  ij = idx[e];
    const float te = t[e];
    const float2 zi = z0[ij.x], zj = z0[ij.y];
    const float2 vi = v0[ij.x], vj = v0[ij.y];
    const float dx = (zi.x - zj.x) + (vi.x - vj.x) * te;
    const float dy = (zi.y - zj.y) + (vi.y - vj.y) * te;
    s = fmaf(dx, dx, fmaf(dy, dy, s));
  }
  const float r = block_reduce(s);
  if (threadIdx.x == 0) partials[blockIdx.x] = r;
}

// ---------------------------------------------------------------------------
// Kernel 2: non-event integral over strict upper-triangular pairs (j > i).
// i broadcast per block (scalar loads), j coalesced. Transcendental-bound.
// ---------------------------------------------------------------------------
__global__ void cvm_pair_kernel(const float2* __restrict__ z0,
                                const float2* __restrict__ v0,
                                int n,
                                const float* __restrict__ t0p,
                                const float* __restrict__ tnp,
                                const float* __restrict__ betap,
                                float* __restrict__ partials) {
  const float t0 = t0p[0];
  const float tn = tnp[0];
  const float bb = betap[0];
  const float half_sqrt_pi = 0.886226925452758014f;   // sqrt(pi)/2

  const int jbase   = blockIdx.x * blockDim.x + threadIdx.x;
  const int jstride = gridDim.x * blockDim.x;

  float s = 0.0f;
  for (int i = blockIdx.y; i < n; i += gridDim.y) {
    const float2 zi = z0[i], vi = v0[i];
    for (int j = i + 1 + jbase; j < n; j += jstride) {
      const float2 zj = z0[j], vj = v0[j];
      const float dzx = zi.x - zj.x, dzy = zi.y - zj.y;
      const float dvx = vi.x - vj.x, dvy = vi.y - vj.y;
      const float a2 = fmaf(dzx, dzx, dzy * dzy);
      const float b2 = fmaf(dvx, dvx, dvy * dvy);
      const float ab = fmaf(dzx, dvx, dzy * dvy);
      const float bnorm = sqrtf(b2);
      const float mu    = ab / b2;
      const float pref  = expf(bb - a2 + ab * mu) * (half_sqrt_pi / bnorm);
      const float integ = pref * (erff(bnorm * (tn + mu)) -
                                  erff(bnorm * (t0 + mu)));
      s += integ;
    }
  }
  const float r = block_reduce(s);
  if (threadIdx.x == 0) partials[blockIdx.y * gridDim.x + blockIdx.x] = r;
}

// ---------------------------------------------------------------------------
// Kernel 3: deterministic final combine on a single block.
// out = n_events * beta - sum(d_event) - sum(integral)
// ---------------------------------------------------------------------------
__global__ void cvm_final_kernel(const float* __restrict__ pairPart, int nPair,
                                 const float* __restrict__ evPart, int nEv,
                                 const float* __restrict__ betap,
                                 int n_events,
                                 float* __restrict__ out) {
  float sp = 0.0f, se = 0.0f;
  for (int k = threadIdx.x; k < nPair; k += blockDim.x) sp += pairPart[k];
  for (int k = threadIdx.x; k < nEv;   k += blockDim.x) se += evPart[k];
  const float rp = block_reduce(sp);
  const float re = block_reduce(se);
  if (threadIdx.x == 0) {
    out[0] = (float)n_events * betap[0] - re - rp;
  }
}

// ---------------------------------------------------------------------------
// Launch.  Inputs (setup_inputs order):
//  0 data_idx int32 (E,2) | 1 data_t f32 (E) | 2 t0 f32[1] | 3 tn f32[1]
//  4 z0 f32 (N,2) | 5 v0 f32 (N,2) | 6 beta f32[1]
// ---------------------------------------------------------------------------
extern "C" void kernel_launch(void* const* d_in, const int* in_sizes, int n_in,
                              void* d_out, int out_size, void* d_ws, size_t ws_size,
                              hipStream_t stream) {
  (void)n_in; (void)out_size; (void)ws_size;
  const int2*   data_idx = (const int2*)  d_in[0];
  const float*  data_t   = (const float*) d_in[1];
  const float*  t0       = (const float*) d_in[2];
  const float*  tn       = (const float*) d_in[3];
  const float2* z0       = (const float2*)d_in[4];
  const float2* v0       = (const float2*)d_in[5];
  const float*  beta     = (const float*) d_in[6];
  float*        out      = (float*)       d_out;

  const int n_events = in_sizes[1];
  const int n_points = in_sizes[4] / 2;

  const int THREADS = 256;

  // Pair grid: x covers one row of j's, y strides over i.
  const int gx = (n_points + THREADS - 1) / THREADS;   // 20 for n=5000
  const int gy = n_points < 256 ? (n_points > 0 ? n_points : 1) : 256;
  const int nPair = gx * gy;                            // 5120 partials

  const int EV_BLOCKS = 1024;

  float* pairPart = (float*)d_ws;            // [0, nPair)
  float* evPart   = pairPart + 8192;         // disjoint region

  cvm_pair_kernel<<<dim3(gx, gy), THREADS, 0, stream>>>(
      z0, v0, n_points, t0, tn, beta, pairPart);

  cvm_event_kernel<<<EV_BLOCKS, THREADS, 0, stream>>>(
      data_idx, data_t, z0, v0, n_events, evPart);

  cvm_final_kernel<<<1, THREADS, 0, stream>>>(
      pairPart, nPair, evPart, EV_BLOCKS, beta, n_events, out);
}